// HGNN_8151847928363
// MI455X (gfx1250) — compile-verified
//
#include <hip/hip_runtime.h>

// ---------------------------------------------------------------------------
// HGNN pipeline for MI455X (gfx1250, wave32).
//  - GEMMs: V_WMMA_F32_16X16X4_F32 (exact fp32 matrix op), LDS-staged A tile.
//  - Edge phase: dst-sorted CSR built once (graph is static across layers),
//    then atomic-free gather (wave per destination node). t/agg are L2
//    resident (25.6 MB each vs 192 MB L2), so this phase runs at L2 rate
//    instead of being limited by 512M float atomics.
// ---------------------------------------------------------------------------

typedef __attribute__((ext_vector_type(2))) float v2f;
typedef __attribute__((ext_vector_type(8))) float v8f;

#define NNODES 50000
#define NEDGES 500000
#define HID    128
#define NREL   4

// ---------------- utility kernels ----------------

__global__ void hgnn_zero_f32(float* __restrict__ p, int n) {
    int i = blockIdx.x * blockDim.x + threadIdx.x;
    if (i < n) p[i] = 0.0f;
}

__global__ void hgnn_zero_i32(int* __restrict__ p, int n) {
    int i = blockIdx.x * blockDim.x + threadIdx.x;
    if (i < n) p[i] = 0;
}

// ---------------- CSR build (once; reused by both layers) -------------------

// one thread per edge: cnt[dst] += 1
__global__ void hgnn_count(const int* __restrict__ dst, int* __restrict__ cnt, int ne) {
    int e = blockIdx.x * blockDim.x + threadIdx.x;
    if (e < ne) atomicAdd(&cnt[dst[e]], 1);
}

// exclusive scan of per-node counts; one block (1024 thr) per relation
__global__ __launch_bounds__(1024) void hgnn_scan(
    const int* __restrict__ cnt, int* __restrict__ off, int* __restrict__ pos, int n)
{
    const int* c = cnt + (size_t)blockIdx.x * n;
    int* o = off + (size_t)blockIdx.x * n;
    int* p = pos + (size_t)blockIdx.x * n;
    __shared__ int part[1024];
    const int tid = threadIdx.x;
    const int chunk = (n + 1023) >> 10;
    const int base = tid * chunk;

    int s = 0;
    for (int i = 0; i < chunk; ++i) {
        int idx = base + i;
        if (idx < n) s += c[idx];
    }
    part[tid] = s;
    __syncthreads();
    // Hillis-Steele inclusive scan over the 1024 partials
    for (int d = 1; d < 1024; d <<= 1) {
        int v = (tid >= d) ? part[tid - d] : 0;
        __syncthreads();
        part[tid] += v;
        __syncthreads();
    }
    int run = (tid == 0) ? 0 : part[tid - 1];
    for (int i = 0; i < chunk; ++i) {
        int idx = base + i;
        if (idx < n) { o[idx] = run; p[idx] = run; run += c[idx]; }
    }
}

// one thread per edge: csr[pos[dst]++] = src
__global__ void hgnn_fill(const int* __restrict__ ei /* [src E][dst E] */,
                          int* __restrict__ pos, int* __restrict__ csr, int ne) {
    int e = blockIdx.x * blockDim.x + threadIdx.x;
    if (e >= ne) return;
    int s = ei[e];
    int d = ei[ne + e];
    int p = atomicAdd(&pos[d], 1);
    csr[p] = s;
}

// dinv = rsqrt(cnt + 1)   (+1 = self loop; always > 0)
__global__ void hgnn_dinv(const int* __restrict__ cnt, float* __restrict__ d, int n) {
    int i = blockIdx.x * blockDim.x + threadIdx.x;
    if (i < n) d[i] = rsqrtf((float)cnt[i] + 1.0f);
}

// out = relu(agg + b[0]+b[1]+b[2]+b[3])  over N*H elements, b is (4,128)
__global__ void hgnn_finalize(const float* __restrict__ agg, const float* __restrict__ b4,
                              float* __restrict__ out, int n) {
    int i = blockIdx.x * blockDim.x + threadIdx.x;
    if (i >= n) return;
    int c = i & 127;
    float v = agg[i] + b4[c] + b4[128 + c] + b4[256 + c] + b4[384 + c];
    out[i] = v > 0.0f ? v : 0.0f;
}

// ---------------- WMMA GEMM: 16 rows/block, 8 waves x 16-col tiles ----------
// MODE 0: out = A@W ;  agg += (A@W) * dinv[row]^2   (GCN self-loop fused)
// MODE 1: A' = A*bn_scale + bn_shift (per column k); out = relu(A'@W + bias)
template <int MODE>
__global__ __launch_bounds__(256) void hgnn_gemm_wmma(
    const float* __restrict__ A, const float* __restrict__ W,
    float* __restrict__ outT, float* __restrict__ agg,
    const float* __restrict__ dinv,
    const float* __restrict__ bnp,   // [0..127]=scale, [128..255]=shift
    const float* __restrict__ bias)  // [128]
{
    __shared__ float As[16][132];    // padded: bank = (r*4 + c) % 64
    __shared__ float sdinv[16];

    const int tid = threadIdx.x;
    const int rowBase = blockIdx.x * 16;

    // cooperative, coalesced load of the 16x128 A tile (8 floats / thread)
    {
        int idx = tid * 8;
        int r = idx >> 7;
        int c = idx & 127;
        const float* src = A + (size_t)(rowBase + r) * HID + c;
#pragma unroll
        for (int i = 0; i < 8; ++i) {
            float v = src[i];
            if (MODE == 1) v = v * bnp[c + i] + bnp[128 + c + i];
            As[r][c + i] = v;
        }
    }
    if (MODE == 0 && tid < 16) sdinv[tid] = dinv[rowBase + tid];
    __syncthreads();

    const int wave = tid >> 5;
    const int lane = tid & 31;
    const int colBase = wave * 16;       // 8 waves -> 8 column tiles of 16
    const int half = lane >> 4;          // 0: K even pair base, 1: +2
    const int l15 = lane & 15;

    v8f acc = {};
    for (int k = 0; k < HID; k += 4) {
        // A 16x4 frag: V0 = K=k (lanes0-15) / K=k+2 (lanes16-31); V1 = +1
        int ka = k + half * 2;
        v2f a, b;
        a.x = As[l15][ka];
        a.y = As[l15][ka + 1];
        // B 4x16 frag, W row-major (K x 128): V0 = row ka, V1 = row ka+1
        const float* wp = W + (size_t)ka * HID + colBase + l15;
        b.x = wp[0];
        b.y = wp[HID];
        acc = __builtin_amdgcn_wmma_f32_16x16x4_f32(
            /*neg_a=*/false, a, /*neg_b=*/false, b,
            /*c_mod=*/(short)0, acc, /*reuse_a=*/false, /*reuse_b=*/false);
    }

    // D layout: VGPR v -> M = v + 8*half, N = l15
    const int col = colBase + l15;
    const int mBase = half * 8;
#pragma unroll
    for (int v = 0; v < 8; ++v) {
        int m = mBase + v;
        size_t off = (size_t)(rowBase + m) * HID + col;
        float val = acc[v];
        if (MODE == 0) {
            outT[off] = val;
            float dv = sdinv[m];
            agg[off] += val * dv * dv;   // tiles disjoint; launches serialize
        } else {
            float r = val + bias[col];
            outT[off] = r > 0.0f ? r : 0.0f;
        }
    }
}

// ---------------- atomic-free edge gather: one wave per dst node ------------
// agg[dst] += sum_{e in CSR[dst]} t[src(e)] * dinv[src]*dinv[dst]
__global__ __launch_bounds__(256) void hgnn_gather(
    const float* __restrict__ t, const int* __restrict__ csr,
    const int* __restrict__ off, const int* __restrict__ cnt,
    const float* __restrict__ dinv, float* __restrict__ agg, int n)
{
    int wid = (blockIdx.x * blockDim.x + threadIdx.x) >> 5;
    int lane = threadIdx.x & 31;
    if (wid >= n) return;
    const int begin = off[wid];
    const int num = cnt[wid];
    const float dd = dinv[wid];
    float ax = 0.0f, ay = 0.0f, az = 0.0f, aw = 0.0f;
    for (int i = 0; i < num; ++i) {
        int s = csr[begin + i];                  // wave-uniform
        float norm = dinv[s] * dd;
        const float4* tp = (const float4*)(t + (size_t)s * HID);
        float4 v = tp[lane];                     // 32 lanes x 16B = full row
        ax += v.x * norm;
        ay += v.y * norm;
        az += v.z * norm;
        aw += v.w * norm;
    }
    float4* ap = (float4*)(agg + (size_t)wid * HID) + lane;
    float4 cur = *ap;                            // dst rows owned exclusively
    cur.x += ax; cur.y += ay; cur.z += az; cur.w += aw;
    *ap = cur;
}

// ---------------- batchnorm statistics --------------------------------------
__global__ __launch_bounds__(128) void hgnn_bn_stats(
    const float* __restrict__ h, float* __restrict__ stats, int nrows)
{
    int c = threadIdx.x;   // 128 columns, one per thread (coalesced rows)
    float s = 0.0f, q = 0.0f;
    for (int r = blockIdx.x; r < nrows; r += gridDim.x) {
        float v = h[(size_t)r * HID + c];
        s += v;
        q += v * v;
    }
    atomicAdd(&stats[c], s);
    atomicAdd(&stats[128 + c], q);
}

__global__ __launch_bounds__(128) void hgnn_bn_finalize(
    const float* __restrict__ stats, const float* __restrict__ gamma,
    const float* __restrict__ beta, float* __restrict__ bnp)
{
    int c = threadIdx.x;
    const float invN = 1.0f / (float)NNODES;
    float mean = stats[c] * invN;
    float var = stats[128 + c] * invN - mean * mean;   // biased variance
    float sc = gamma[c] * rsqrtf(var + 1e-5f);
    bnp[c] = sc;
    bnp[128 + c] = beta[c] - mean * sc;
}

// ---------------- final 128->2 projection: one wave per node ----------------
__global__ __launch_bounds__(256) void hgnn_lin2(
    const float* __restrict__ h, const float* __restrict__ W /* (128,2) */,
    const float* __restrict__ b, float* __restrict__ out, int nrows)
{
    int wid = (blockIdx.x * blockDim.x + threadIdx.x) >> 5;
    int lane = threadIdx.x & 31;
    if (wid >= nrows) return;
    const float* hp = h + (size_t)wid * HID;
    float p0 = 0.0f, p1 = 0.0f;
#pragma unroll
    for (int j = 0; j < 4; ++j) {
        int k = lane + 32 * j;
        float v = hp[k];
        p0 += v * W[k * 2 + 0];
        p1 += v * W[k * 2 + 1];
    }
#pragma unroll
    for (int o = 16; o > 0; o >>= 1) {
        p0 += __shfl_down(p0, o, 32);
        p1 += __shfl_down(p1, o, 32);
    }
    if (lane == 0) {
        out[(size_t)wid * 2 + 0] = p0 + b[0];
        out[(size_t)wid * 2 + 1] = p1 + b[1];
    }
}

// ---------------------------------------------------------------------------
extern "C" void kernel_launch(void* const* d_in, const int* in_sizes, int n_in,
                              void* d_out, int out_size, void* d_ws, size_t ws_size,
                              hipStream_t stream) {
    const float* x     = (const float*)d_in[0];
    const int*   ei    = (const int*)d_in[1];     // (4, 2, E)
    const float* W1    = (const float*)d_in[2];   // (4,128,128)
    const float* b1    = (const float*)d_in[3];   // (4,128)
    const float* W2    = (const float*)d_in[4];
    const float* b2    = (const float*)d_in[5];
    const float* gamma = (const float*)d_in[6];
    const float* beta  = (const float*)d_in[7];
    const float* l1W   = (const float*)d_in[8];
    const float* l1b   = (const float*)d_in[9];
    const float* l2W   = (const float*)d_in[10];
    const float* l2b   = (const float*)d_in[11];
    float* out = (float*)d_out;

    const int N = NNODES, E = NEDGES;
    const size_t NH = (size_t)N * HID;

    // -------- workspace layout --------
    float* fws   = (float*)d_ws;
    float* t     = fws;                    // N*H  (GEMM scratch, later h3)
    float* agg   = fws + NH;               // N*H  (message accumulator)
    float* h     = fws + 2 * NH;           // N*H  (h1, then h2 in place)
    float* dinv  = fws + 3 * NH;           // 4*N
    float* stats = dinv + 4 * (size_t)N;   // 256
    float* bnp   = stats + 256;            // 256
    int*   iws   = (int*)(bnp + 256);
    int*   cnt   = iws;                    // 4*N  in-degrees
    int*   off   = iws + 4 * (size_t)N;    // 4*N  CSR row offsets
    int*   pos   = iws + 8 * (size_t)N;    // 4*N  fill cursors
    int*   csr   = iws + 12 * (size_t)N;   // 4*E  src ids sorted by dst

    const int gNH   = (int)((NH + 255) / 256);
    const int g4N   = (4 * N + 255) / 256;
    const int gE    = (E + 255) / 256;
    const int gGemm = N / 16;   // 50000 / 16 = 3125 exactly
    const int gNw   = (int)(((size_t)N * 32 + 255) / 256);   // wave per node

    // --- build dst-sorted CSR per relation (reused by both layers) + dinv
    hgnn_zero_i32<<<g4N, 256, 0, stream>>>(cnt, 4 * N);
    for (int r = 0; r < NREL; ++r)
        hgnn_count<<<gE, 256, 0, stream>>>(ei + (size_t)r * 2 * E + E, cnt + (size_t)r * N, E);
    hgnn_scan<<<NREL, 1024, 0, stream>>>(cnt, off, pos, N);
    for (int r = 0; r < NREL; ++r)
        hgnn_fill<<<gE, 256, 0, stream>>>(ei + (size_t)r * 2 * E, pos + (size_t)r * N,
                                          csr + (size_t)r * E, E);
    hgnn_dinv<<<g4N, 256, 0, stream>>>(cnt, dinv, 4 * N);

    // --- layer 1
    hgnn_zero_f32<<<gNH, 256, 0, stream>>>(agg, (int)NH);
    for (int r = 0; r < NREL; ++r) {
        hgnn_gemm_wmma<0><<<gGemm, 256, 0, stream>>>(
            x, W1 + (size_t)r * HID * HID, t, agg, dinv + (size_t)r * N, nullptr, nullptr);
        hgnn_gather<<<gNw, 256, 0, stream>>>(t, csr + (size_t)r * E, off + (size_t)r * N,
                                             cnt + (size_t)r * N, dinv + (size_t)r * N, agg, N);
    }
    hgnn_finalize<<<gNH, 256, 0, stream>>>(agg, b1, h, (int)NH);

    // --- layer 2 (h overwritten in place by finalize afterwards)
    hgnn_zero_f32<<<gNH, 256, 0, stream>>>(agg, (int)NH);
    for (int r = 0; r < NREL; ++r) {
        hgnn_gemm_wmma<0><<<gGemm, 256, 0, stream>>>(
            h, W2 + (size_t)r * HID * HID, t, agg, dinv + (size_t)r * N, nullptr, nullptr);
        hgnn_gather<<<gNw, 256, 0, stream>>>(t, csr + (size_t)r * E, off + (size_t)r * N,
                                             cnt + (size_t)r * N, dinv + (size_t)r * N, agg, N);
    }
    hgnn_finalize<<<gNH, 256, 0, stream>>>(agg, b2, h, (int)NH);

    // --- batchnorm stats -> scale/shift
    hgnn_zero_f32<<<1, 256, 0, stream>>>(stats, 256);
    hgnn_bn_stats<<<512, 128, 0, stream>>>(h, stats, N);
    hgnn_bn_finalize<<<1, 128, 0, stream>>>(stats, gamma, beta, bnp);

    // --- lin1 with BN fused on input, relu+bias epilogue -> t
    hgnn_gemm_wmma<1><<<gGemm, 256, 0, stream>>>(h, l1W, t, nullptr, nullptr, bnp, l1b);

    // --- lin2 (128 -> 2)
    hgnn_lin2<<<gNw, 256, 0, stream>>>(t, l2W, l2b, out, N);
}